// ECRF_30545807409306
// MI455X (gfx1250) — compile-verified
//
#include <hip/hip_runtime.h>
#include <math.h>

#define B_   2
#define CF_  64
#define H_   256
#define W_   256
#define CH_  32
#define HW_  (H_ * W_)
#define IMGH 512
#define IMGW 512

typedef __attribute__((ext_vector_type(2))) float v2f;
typedef __attribute__((ext_vector_type(8))) float v8f;

// Sinusoidal basis value pos[cc, (y,x)]:
//   cc in [0,32): x-encoding; cc in [32,64): y-encoding
//   pair = (cc&31)>>1; div = 10000^(-pair/16); even->sin, odd->cos
// Uses hardware v_sin/v_cos (args bounded by 255 rad -> fast path is accurate).
__device__ __forceinline__ float pos_val(int cc, int x, int y) {
    const int   cc2   = cc & 31;
    const float coord = (cc < 32) ? (float)x : (float)y;
    const int   pair  = cc2 >> 1;
    // 10000^{-pair/16} = exp2(-pair * log2(10000)/16)
    const float dv  = exp2f(-0.83048202372185f * (float)pair);
    const float ang = coord * dv;
    return (cc2 & 1) ? __cosf(ang) : __sinf(ang);
}

// ---------------------------------------------------------------------------
// Kernel 1: pe[o, p] = sum_cc w_pos[o, cc] * pos[cc, p]   (M=32, K=64, N=65536)
// One wave computes a 16(M) x 16(N) f32 tile via 16 chained V_WMMA_F32_16X16X4_F32.
// A layout (16x4 f32): lanes 0-15 hold M=lane, K={k0,k0+1}; lanes 16-31 K={k0+2,k0+3}.
// B layout (4x16 f32): mirrored (lane%16 = N column, lane/16 selects K pair).
// C/D layout (16x16 f32, 8 VGPRs): vgpr r, lanes 0-15 -> M=r, lanes 16-31 -> M=r+8.
// ---------------------------------------------------------------------------
__global__ void pos_gemm_wmma(const float* __restrict__ w_pos,
                              float* __restrict__ pe) {
    const int lane  = threadIdx.x & 31;
    const int wave  = threadIdx.x >> 5;          // 0..7
    const int otile = wave & 1;                  // o base 0 / 16
    const int ptile = wave >> 1;                 // 0..3
    const int pix0  = blockIdx.x * 64 + ptile * 16;   // 16 consecutive pixels, same row
    const int y     = pix0 >> 8;                 // W_ == 256
    const int x0    = pix0 & 255;

    const int mrow  = lane & 15;                 // A row (M) / B column (N)
    const int khalf = (lane >> 4) << 1;          // 0 or 2 (K sub-pair)
    const int xpix  = x0 + mrow;

    v8f acc = {};
#pragma unroll
    for (int k0 = 0; k0 < 64; k0 += 4) {
        const int ka = k0 + khalf;
        v2f a, bb;
        a.x  = w_pos[(otile * 16 + mrow) * 64 + ka];
        a.y  = w_pos[(otile * 16 + mrow) * 64 + ka + 1];
        bb.x = pos_val(ka,     xpix, y);
        bb.y = pos_val(ka + 1, xpix, y);
        acc = __builtin_amdgcn_wmma_f32_16x16x4_f32(
            /*neg_a=*/false, a, /*neg_b=*/false, bb,
            /*c_mod=*/(short)0, acc, /*reuse_a=*/false, /*reuse_b=*/false);
    }

    const int nhalf = lane >> 4;                 // M offset 0 / 8
    const int ncol  = lane & 15;
#pragma unroll
    for (int r = 0; r < 8; ++r) {
        const int o = otile * 16 + nhalf * 8 + r;
        pe[o * HW_ + pix0 + ncol] = acc[r];
    }
}

// ---------------------------------------------------------------------------
// Kernel 2: em[b, 0:32, p]  = w_img (3->32) applied to antialiased 2x downsample
//           em[b, 32:64, p] = pe[:, p]
// jax.image.resize(bilinear, antialias=True), scale 0.5: taps at 2i-1..2i+2,
// per-dim weights [1,3,3,1]/8, renormalized per dimension over in-range taps.
// ---------------------------------------------------------------------------
__global__ void build_em_kernel(const float* __restrict__ img,
                                const float* __restrict__ w_img,
                                const float* __restrict__ pe,
                                float* __restrict__ em) {
    const int idx = blockIdx.x * blockDim.x + threadIdx.x;   // over B*HW
    if (idx >= B_ * HW_) return;
    const int b = idx / HW_;
    const int p = idx - b * HW_;
    const int y = p >> 8;
    const int x = p & 255;

    const float tw[4] = {1.f, 3.f, 3.f, 1.f};
    float wy[4], wx[4];
    int   jy[4], jx[4];
    float sy = 0.f, sx = 0.f;
#pragma unroll
    for (int t = 0; t < 4; ++t) {
        jy[t] = 2 * y - 1 + t;
        jx[t] = 2 * x - 1 + t;
        wy[t] = ((unsigned)jy[t] < (unsigned)IMGH) ? tw[t] : 0.f;
        wx[t] = ((unsigned)jx[t] < (unsigned)IMGW) ? tw[t] : 0.f;
        sy += wy[t];
        sx += wx[t];
    }
    const float inv = 1.0f / (sy * sx);

    float r[3];
#pragma unroll
    for (int c = 0; c < 3; ++c) {
        const float* ib  = img + ((long)b * 3 + c) * (long)(IMGH * IMGW);
        float        acc = 0.f;
#pragma unroll
        for (int ty = 0; ty < 4; ++ty) {
            if (wy[ty] != 0.f) {
                const float* row  = ib + (long)jy[ty] * IMGW;
                float        accx = 0.f;
#pragma unroll
                for (int tx = 0; tx < 4; ++tx) {
                    if (wx[tx] != 0.f) accx = fmaf(wx[tx], row[jx[tx]], accx);
                }
                acc = fmaf(wy[ty], accx, acc);
            }
        }
        r[c] = acc * inv;
    }

    float* emb = em + (long)b * 64 * HW_;
#pragma unroll
    for (int c = 0; c < 32; ++c) {
        const float v = fmaf(w_img[c * 3 + 0], r[0],
                        fmaf(w_img[c * 3 + 1], r[1],
                             w_img[c * 3 + 2] * r[2]));
        emb[c * HW_ + p] = v;
    }
#pragma unroll
    for (int c = 0; c < 32; ++c) {
        emb[(32 + c) * HW_ + p] = pe[c * HW_ + p];
    }
}

// ---------------------------------------------------------------------------
// Kernel 3: phi[b, k, p] = sigmoid( <w_comp, inp(p+dk)> ) * <em(p), em(p+dk)>
// 9 dilated (+-2) neighbors, zero-padded. Out-of-range -> both dots 0 -> phi 0.
// ---------------------------------------------------------------------------
__global__ void phi_kernel(const float* __restrict__ em,
                           const float* __restrict__ inp,
                           const float* __restrict__ w_comp,
                           float* __restrict__ phi) {
    const int idx = blockIdx.x * blockDim.x + threadIdx.x;   // over B*HW
    if (idx >= B_ * HW_) return;
    const int b = idx / HW_;
    const int p = idx - b * HW_;
    const int y = p >> 8;
    const int x = p & 255;

    int  off[9];
    bool valid[9];
#pragma unroll
    for (int k = 0; k < 9; ++k) {
        const int dy = (k / 3 - 1) * 2;
        const int dx = (k % 3 - 1) * 2;
        valid[k] = ((unsigned)(y + dy) < (unsigned)H_) &&
                   ((unsigned)(x + dx) < (unsigned)W_);
        off[k]   = dy * W_ + dx;
    }

    const float* emb = em  + (long)b * 64 * HW_ + p;
    const float* inb = inp + (long)b * 64 * HW_ + p;
    float fs[9] = {}, cp[9] = {};
    for (int c = 0; c < 64; ++c) {
        const float  ec = emb[(long)c * HW_];
        const float  wc = w_comp[c];
        const float* ep = emb + (long)c * HW_;
        const float* ip = inb + (long)c * HW_;
#pragma unroll
        for (int k = 0; k < 9; ++k) {
            if (valid[k]) {
                fs[k] = fmaf(ec, ep[off[k]], fs[k]);
                cp[k] = fmaf(wc, ip[off[k]], cp[k]);
            }
        }
    }

    float* ph = phi + (long)b * 9 * HW_ + p;
#pragma unroll
    for (int k = 0; k < 9; ++k) {
        const float sg = 1.0f / (1.0f + __expf(-cp[k]));
        ph[(long)k * HW_] = fs[k] * sg;          // invalid k: fs==0 -> 0
    }
}

// ---------------------------------------------------------------------------
// Kernel 4: out[b, c, p] = sum_k inp[b, c, p+dk] * phi[b, k, p]
// ---------------------------------------------------------------------------
__global__ void gather_kernel(const float* __restrict__ inp,
                              const float* __restrict__ phi,
                              float* __restrict__ out) {
    const long idx = (long)blockIdx.x * blockDim.x + threadIdx.x;  // B*CF*HW
    if (idx >= (long)B_ * CF_ * HW_) return;
    const int  p  = (int)(idx & (HW_ - 1));      // HW = 2^16
    const long bc = idx >> 16;
    const int  b  = (int)(bc >> 6);              // CF = 64
    const int  y  = p >> 8;
    const int  x  = p & 255;

    const float* ph = phi + (long)b * 9 * HW_ + p;
    const float* ip = inp + bc * HW_ + p;
    float acc = 0.f;
#pragma unroll
    for (int k = 0; k < 9; ++k) {
        const int dy = (k / 3 - 1) * 2;
        const int dx = (k % 3 - 1) * 2;
        if (((unsigned)(y + dy) < (unsigned)H_) &&
            ((unsigned)(x + dx) < (unsigned)W_)) {
            acc = fmaf(ip[dy * W_ + dx], ph[(long)k * HW_], acc);
        }
    }
    out[idx] = acc;
}

// ---------------------------------------------------------------------------
// Host-side launch. Inputs: inp[B,64,256,256], img[B,3,512,512],
// w_pos[32,64], w_img[32,3], w_comp[64]; out[B,64,256,256] f32.
// Workspace layout (floats): pe[32*HW] | em[B*64*HW] | phi[B*9*HW] ~= 44.5 MB.
// ---------------------------------------------------------------------------
extern "C" void kernel_launch(void* const* d_in, const int* in_sizes, int n_in,
                              void* d_out, int out_size, void* d_ws, size_t ws_size,
                              hipStream_t stream) {
    const float* inp    = (const float*)d_in[0];
    const float* img    = (const float*)d_in[1];
    const float* w_pos  = (const float*)d_in[2];
    const float* w_img  = (const float*)d_in[3];
    const float* w_comp = (const float*)d_in[4];
    float*       out    = (float*)d_out;

    float* pe  = (float*)d_ws;                       // 32 * HW
    float* em  = pe + (size_t)CH_ * HW_;             // B * 64 * HW
    float* phi = em + (size_t)B_ * 64 * HW_;         // B * 9 * HW

    // 1) positional-embedding GEMM (WMMA): 1024 blocks x 8 waves,
    //    each wave one 16x16 output tile, K=64.
    pos_gemm_wmma<<<HW_ / 64, 256, 0, stream>>>(w_pos, pe);

    // 2) em = [w_img * resize(img) || pe]
    build_em_kernel<<<(B_ * HW_ + 255) / 256, 256, 0, stream>>>(img, w_img, pe, em);

    // 3) phi = sigmoid(comp) * feature_score
    phi_kernel<<<(B_ * HW_ + 255) / 256, 256, 0, stream>>>(em, inp, w_comp, phi);

    // 4) out = sum_k inp(neighbor_k) * phi_k
    gather_kernel<<<(int)(((long)B_ * CF_ * HW_ + 255) / 256), 256, 0, stream>>>(
        inp, phi, out);
}